// MessagePassing_StaticDiag_65274912965246
// MI455X (gfx1250) — compile-verified
//
#include <hip/hip_runtime.h>
#include <hip/hip_bf16.h>

// ---------------------------------------------------------------------------
// Message-passing GNN, 3 rounds. C=16 channels, H=64 hidden.
// V_WMMA_F32_16X16X4_F32 with register-resident weights; 32-bit (int) flat
// indexing so loads/stores use SGPR-base + 32-bit voffset addressing.
// ---------------------------------------------------------------------------

#define CH 16
#define HID 64
#define LSTRIDE 68   // dwords per lane-column in LDS (4-dword bank skew, 16B aligned)

typedef float v2f __attribute__((ext_vector_type(2)));
typedef float v4f __attribute__((ext_vector_type(4)));
typedef float v8f __attribute__((ext_vector_type(8)));

__device__ __forceinline__ v8f wmma4(v2f a, v2f b, v8f c) {
    // (neg_a, A, neg_b, B, c_mod, C, reuse_a, reuse_b)
    return __builtin_amdgcn_wmma_f32_16x16x4_f32(false, a, false, b, (short)0, c,
                                                 false, false);
}

// --------------------------- zero scratch ----------------------------------
__global__ void zero_kernel(float* __restrict__ p, int n) {
    int i = blockIdx.x * 256 + threadIdx.x;
    if (i < n) p[i] = 0.0f;
}

// ------------------- gather + segment-sum (atomics) ------------------------
// sent[c, s[e]] += edges[c, e] * nodes[c, r[e]]   for all e (incl. diagonal)
__global__ void gather_scatter_kernel(const float* __restrict__ nodes,
                                      const float* __restrict__ edges,
                                      const int* __restrict__ senders,
                                      const int* __restrict__ receivers,
                                      float* __restrict__ sent,
                                      int N, int E) {
    int e = blockIdx.x * 256 + threadIdx.x;
    if (e >= E) return;
    int s = senders[e];
    int r = receivers[e];
#pragma unroll
    for (int c = 0; c < CH; ++c) {
        float v = edges[c * E + e] * nodes[c * N + r];
        unsafeAtomicAdd(&sent[c * N + s], v);
    }
}

// --------------------------- node MLP --------------------------------------
// X = concat(nodes, sent) [32 x N];  out = W2 @ relu(W1 @ X + b1) + b2
__global__ __launch_bounds__(128) void node_mlp_kernel(
        const float* __restrict__ nodes, const float* __restrict__ sent,
        const float* __restrict__ W1, const float* __restrict__ b1,
        const float* __restrict__ W2, const float* __restrict__ b2,
        float* __restrict__ out, int N) {
    __shared__ float lds_h[4][16 * LSTRIDE];

    int wave = threadIdx.x >> 5;
    int lane = threadIdx.x & 31;
    int lm   = lane & 15;          // M (A rows) / N (B cols) index
    int hi   = lane >> 4;          // lane group
    int kg   = hi << 1;            // K sub-offset for this lane group
    float* hl = lds_h[wave] + lm * LSTRIDE;   // this lane's hidden column

    // ---- hoist all weight fragments + bias-initialized accumulators ----
    v2f a1[8][4];                  // W1 (64x32): 8 K-steps x 4 row tiles
#pragma unroll
    for (int ks = 0; ks < 8; ++ks) {
        int klo = ks * 4 + kg;
#pragma unroll
        for (int rt = 0; rt < 4; ++rt) {
            int m = rt * 16 + lm;
            a1[ks][rt].x = W1[m * 32 + klo];
            a1[ks][rt].y = W1[m * 32 + klo + 1];
        }
    }
    v2f a2[16];                    // W2 (16x64): 16 K-steps
#pragma unroll
    for (int ks = 0; ks < 16; ++ks) {
        int klo = ks * 4 + kg;
        a2[ks].x = W2[lm * 64 + klo];
        a2[ks].y = W2[lm * 64 + klo + 1];
    }
    v8f binit[4];
#pragma unroll
    for (int rt = 0; rt < 4; ++rt)
#pragma unroll
        for (int v = 0; v < 8; ++v) binit[rt][v] = b1[rt * 16 + hi * 8 + v];
    v8f b2init;
#pragma unroll
    for (int v = 0; v < 8; ++v) b2init[v] = b2[hi * 8 + v];

    int ntiles = N >> 4;
    int totalWaves = (int)gridDim.x * 4;
    for (int tile = (int)blockIdx.x * 4 + wave; tile < ntiles; tile += totalWaves) {
        int col = (tile << 4) + lm;

        // layer 1: (64x32) @ (32x16)
        v8f acc[4];
#pragma unroll
        for (int rt = 0; rt < 4; ++rt) acc[rt] = binit[rt];
#pragma unroll
        for (int ks = 0; ks < 8; ++ks) {
            int klo = ks * 4 + kg;
            const float* xb = (klo < 16) ? nodes : sent;
            int kk = klo & 15;
            v2f bfr;
            bfr.x = xb[kk * N + col];
            bfr.y = xb[(kk + 1) * N + col];
#pragma unroll
            for (int rt = 0; rt < 4; ++rt) acc[rt] = wmma4(a1[ks][rt], bfr, acc[rt]);
        }

        // relu -> LDS (column-major per lane, two b128 stores per row tile)
#pragma unroll
        for (int rt = 0; rt < 4; ++rt) {
            int row = rt * 16 + hi * 8;
            v4f lo, hi4;
#pragma unroll
            for (int v = 0; v < 4; ++v) lo[v] = fmaxf(acc[rt][v], 0.0f);
#pragma unroll
            for (int v = 0; v < 4; ++v) hi4[v] = fmaxf(acc[rt][v + 4], 0.0f);
            *(v4f*)&hl[row]     = lo;
            *(v4f*)&hl[row + 4] = hi4;
        }
        __builtin_amdgcn_wave_barrier();

        // layer 2: (16x64) @ (64x16)
        v8f acc2 = b2init;
#pragma unroll
        for (int ks = 0; ks < 16; ++ks) {
            int klo = ks * 4 + kg;
            v2f bfr = *(v2f*)&hl[klo];
            acc2 = wmma4(a2[ks], bfr, acc2);
        }
#pragma unroll
        for (int v = 0; v < 8; ++v) {
            int row = v + hi * 8;
            out[row * N + col] = acc2[v];
        }
        __builtin_amdgcn_wave_barrier();
    }
}

// --------------------------- edge MLP --------------------------------------
// feat = [edges[:,e]; nodes[:,s]; nodes[:,r]] (48) -> 64 relu -> 16, in place.
// Off-diagonal edges are exactly indices N..E-1.
__global__ __launch_bounds__(128) void edge_mlp_kernel(
        float* __restrict__ edges, const float* __restrict__ nodes,
        const int* __restrict__ senders, const int* __restrict__ receivers,
        const float* __restrict__ W1, const float* __restrict__ b1,
        const float* __restrict__ W2, const float* __restrict__ b2,
        int N, int E, int Eoff) {
    __shared__ float lds_h[4][16 * LSTRIDE];

    int wave = threadIdx.x >> 5;
    int lane = threadIdx.x & 31;
    int lm   = lane & 15;
    int hi   = lane >> 4;
    int kg   = hi << 1;
    float* hl = lds_h[wave] + lm * LSTRIDE;

    // ---- hoist weight fragments + bias accumulator inits ----
    v2f a1[12][4];                 // W1 (64x48): 12 K-steps x 4 row tiles
#pragma unroll
    for (int ks = 0; ks < 12; ++ks) {
        int klo = ks * 4 + kg;
#pragma unroll
        for (int rt = 0; rt < 4; ++rt) {
            int m = rt * 16 + lm;
            a1[ks][rt].x = W1[m * 48 + klo];
            a1[ks][rt].y = W1[m * 48 + klo + 1];
        }
    }
    v2f a2[16];                    // W2 (16x64)
#pragma unroll
    for (int ks = 0; ks < 16; ++ks) {
        int klo = ks * 4 + kg;
        a2[ks].x = W2[lm * 64 + klo];
        a2[ks].y = W2[lm * 64 + klo + 1];
    }
    v8f binit[4];
#pragma unroll
    for (int rt = 0; rt < 4; ++rt)
#pragma unroll
        for (int v = 0; v < 8; ++v) binit[rt][v] = b1[rt * 16 + hi * 8 + v];
    v8f b2init;
#pragma unroll
    for (int v = 0; v < 8; ++v) b2init[v] = b2[hi * 8 + v];

    int ntiles = Eoff >> 4;
    int totalWaves = (int)gridDim.x * 4;
    for (int tile = (int)blockIdx.x * 4 + wave; tile < ntiles; tile += totalWaves) {
        int e = N + (tile << 4) + lm;
        int s = senders[e];
        int r = receivers[e];

        // layer 1: (64x48) @ (48x16)
        v8f acc[4];
#pragma unroll
        for (int rt = 0; rt < 4; ++rt) acc[rt] = binit[rt];
#pragma unroll
        for (int ks = 0; ks < 12; ++ks) {
            int klo = ks * 4 + kg;
            int sec = klo >> 4;          // 0: edge, 1: nodes[s], 2: nodes[r]
            int kk = klo & 15;
            v2f bfr;
            if (sec == 0) {
                bfr.x = edges[kk * E + e];
                bfr.y = edges[(kk + 1) * E + e];
            } else {
                int idx = (sec == 1) ? s : r;
                bfr.x = nodes[kk * N + idx];
                bfr.y = nodes[(kk + 1) * N + idx];
            }
#pragma unroll
            for (int rt = 0; rt < 4; ++rt) acc[rt] = wmma4(a1[ks][rt], bfr, acc[rt]);
        }

        // relu -> LDS
#pragma unroll
        for (int rt = 0; rt < 4; ++rt) {
            int row = rt * 16 + hi * 8;
            v4f lo, hi4;
#pragma unroll
            for (int v = 0; v < 4; ++v) lo[v] = fmaxf(acc[rt][v], 0.0f);
#pragma unroll
            for (int v = 0; v < 4; ++v) hi4[v] = fmaxf(acc[rt][v + 4], 0.0f);
            *(v4f*)&hl[row]     = lo;
            *(v4f*)&hl[row + 4] = hi4;
        }
        __builtin_amdgcn_wave_barrier();

        // layer 2: (16x64) @ (64x16)
        v8f acc2 = b2init;
#pragma unroll
        for (int ks = 0; ks < 16; ++ks) {
            int klo = ks * 4 + kg;
            v2f bfr = *(v2f*)&hl[klo];
            acc2 = wmma4(a2[ks], bfr, acc2);
        }
#pragma unroll
        for (int v = 0; v < 8; ++v) {
            int row = v + hi * 8;
            edges[row * E + e] = acc2[v];
        }
        __builtin_amdgcn_wave_barrier();
    }
}

// ---------------------------------------------------------------------------
extern "C" void kernel_launch(void* const* d_in, const int* in_sizes, int n_in,
                              void* d_out, int out_size, void* d_ws, size_t ws_size,
                              hipStream_t stream) {
    const float* nodes_in = (const float*)d_in[0];
    const float* edges_in = (const float*)d_in[1];
    const int*   senders   = (const int*)d_in[2];
    const int*   receivers = (const int*)d_in[3];
    const float* W1n = (const float*)d_in[4];
    const float* b1n = (const float*)d_in[5];
    const float* W2n = (const float*)d_in[6];
    const float* b2n = (const float*)d_in[7];
    const float* W1e = (const float*)d_in[8];
    const float* b1e = (const float*)d_in[9];
    const float* W2e = (const float*)d_in[10];
    const float* b2e = (const float*)d_in[11];
    const int mp_rounds = 3;  // reference setup value

    int N = in_sizes[0] / CH;
    int E = in_sizes[2];
    int Eoff = E - N;

    // d_out layout: nodes [CH*N] | edges [CH*E] | senders [E] | receivers [E]
    float* out_nodes = (float*)d_out;
    float* out_edges = out_nodes + (size_t)CH * N;
    float* out_send  = out_edges + (size_t)CH * E;
    float* out_recv  = out_send + (size_t)E;

    // workspace: sent [CH*N] | nodes_tmp [CH*N]
    float* sent = (float*)d_ws;
    float* ntmp = sent + (size_t)CH * N;

    // working edges live in d_out; re-seeded every launch (deterministic)
    hipMemcpyAsync(out_edges, edges_in, sizeof(float) * (size_t)CH * E,
                   hipMemcpyDeviceToDevice, stream);
    hipMemcpyAsync(out_send, senders, sizeof(int) * (size_t)E,
                   hipMemcpyDeviceToDevice, stream);
    hipMemcpyAsync(out_recv, receivers, sizeof(int) * (size_t)E,
                   hipMemcpyDeviceToDevice, stream);

    // node ping-pong: in -> out_nodes -> ntmp -> out_nodes (ends in d_out)
    const float* cur[3] = {nodes_in, out_nodes, ntmp};
    float*       nxt[3] = {out_nodes, ntmp, out_nodes};

    int sent_elems = CH * N;
    int zero_blocks = (sent_elems + 255) / 256;
    int gs_blocks = (E + 255) / 256;

    // ~8 tiles per wave (node), ~16 tiles per wave (edge); 4 waves / block
    int node_tiles = N >> 4;
    int edge_tiles = Eoff >> 4;
    int node_blocks = ((node_tiles + 7) / 8 + 3) / 4;
    int edge_blocks = ((edge_tiles + 15) / 16 + 3) / 4;
    if (node_blocks < 1) node_blocks = 1;
    if (edge_blocks < 1) edge_blocks = 1;

    for (int rd = 0; rd < mp_rounds; ++rd) {
        zero_kernel<<<zero_blocks, 256, 0, stream>>>(sent, sent_elems);
        gather_scatter_kernel<<<gs_blocks, 256, 0, stream>>>(
            cur[rd], out_edges, senders, receivers, sent, N, E);
        node_mlp_kernel<<<node_blocks, 128, 0, stream>>>(
            cur[rd], sent, W1n, b1n, W2n, b2n, nxt[rd], N);
        edge_mlp_kernel<<<edge_blocks, 128, 0, stream>>>(
            out_edges, nxt[rd], senders, receivers, W1e, b1e, W2e, b2e,
            N, E, Eoff);
    }
}